// GlobalGatedUpdate_26036091749094
// MI455X (gfx1250) — compile-verified
//
#include <hip/hip_runtime.h>

typedef __attribute__((ext_vector_type(2))) float v2f;
typedef __attribute__((ext_vector_type(4))) float v4f;
typedef __attribute__((ext_vector_type(8))) float v8f;

constexpr int BATCH = 32;
constexpr int NODES = 64;     // nodes per graph
constexpr int ITEMS = 50000;  // I
constexpr int EMB   = 64;     // F

// ---------------------------------------------------------------------------
// Kernel A: out[b,i,:] = item_emb[i,:] for all (b,i).  410 MB of streaming
// stores -> the entire runtime.  128-bit NT stores; item_emb (12.8 MB) stays
// L2-resident across the 32 graph slabs.
// ---------------------------------------------------------------------------
__global__ void broadcast_emb_kernel(const v4f* __restrict__ emb4,
                                     v4f* __restrict__ out4) {
    const int per_graph = ITEMS * EMB / 4;  // 800000 float4 per graph
    int idx = blockIdx.x * blockDim.x + threadIdx.x;
    if (idx < per_graph) {
        v4f v = emb4[idx];
        __builtin_nontemporal_store(v, &out4[(size_t)blockIdx.y * per_graph + idx]);
    }
}

// ---------------------------------------------------------------------------
// Kernel B: per-graph segment mean via WMMA membership matmul, then gated
// blend written over the broadcast output for touched (graph,item) rows.
//   S = M * Ffeat,  M[j,k] = (id[j]==id[k]) ? 1 : 0   (64x64x64 f32 GEMM)
// One block per graph, 512 threads = 16 waves, one 16x16 tile per wave,
// 16 steps of V_WMMA_F32_16X16X4_F32.
// ---------------------------------------------------------------------------
__global__ void gated_update_kernel(const int*   __restrict__ nodes,
                                    const float* __restrict__ nodes_output,
                                    const float* __restrict__ alpha,
                                    const float* __restrict__ item_emb,
                                    float*       __restrict__ out) {
    __shared__ int   s_id[NODES];
    __shared__ float s_feat[NODES][EMB];
    __shared__ float s_cnt[NODES];
    __shared__ int   s_rep[NODES];

    const int g   = blockIdx.x;
    const int tid = threadIdx.x;  // 0..511

    if (tid < NODES) s_id[tid] = nodes[g * NODES + tid];
    {   // stage this graph's node features into LDS (1024 x float4)
        const v4f* src = (const v4f*)(nodes_output + (size_t)g * NODES * EMB);
        v4f* dst = (v4f*)&s_feat[0][0];
        for (int t = tid; t < NODES * EMB / 4; t += blockDim.x) dst[t] = src[t];
    }
    __syncthreads();

    if (tid < NODES) {  // occurrence count + first-occurrence representative
        const int id = s_id[tid];
        int c = 0, rep = 1;
        for (int k = 0; k < NODES; ++k) {
            const bool eq = (s_id[k] == id);
            c += eq ? 1 : 0;
            if (eq && k < tid) rep = 0;
        }
        s_cnt[tid] = (float)c;
        s_rep[tid] = rep;
    }
    __syncthreads();

    const int wave = tid >> 5;
    const int lane = tid & 31;
    const int lh   = lane >> 4;   // lane half: 0 or 1
    const int l    = lane & 15;
    const int m0   = (wave >> 2) * 16;  // output node-tile
    const int n0   = (wave & 3) * 16;   // output feature-tile

    // A-fragment row for this lane (both lane halves carry M = m0..m0+15)
    const int idj = s_id[m0 + l];

    v8f acc = {};
#pragma unroll
    for (int kc = 0; kc < 16; ++kc) {
        const int k0 = kc * 4 + 2 * lh;  // lanes 0-15: K=k0,k0+1 ; lanes 16-31: K=k0+2,k0+3
        v2f a, b;
        a.x = (s_id[k0]     == idj) ? 1.0f : 0.0f;
        a.y = (s_id[k0 + 1] == idj) ? 1.0f : 0.0f;
        b.x = s_feat[k0][n0 + l];
        b.y = s_feat[k0 + 1][n0 + l];
        acc = __builtin_amdgcn_wmma_f32_16x16x4_f32(
            /*neg_a=*/false, a, /*neg_b=*/false, b,
            /*c_mod=*/(short)0, acc, /*reuse_a=*/false, /*reuse_b=*/false);
    }

    // C/D layout: VGPR r -> node row m0 + r + 8*lh, feature col n0 + l.
    // Representative lanes overwrite the broadcast rows with the gated blend.
#pragma unroll
    for (int r = 0; r < 8; ++r) {
        const int j = m0 + r + 8 * lh;
        if (s_rep[j]) {
            const int   id  = s_id[j];
            const float a   = alpha[id];
            const int   f   = n0 + l;
            const float avg = acc[r] / s_cnt[j];
            const float e   = item_emb[(size_t)id * EMB + f];
            out[((size_t)g * ITEMS + id) * EMB + f] = (1.0f - a) * e + a * avg;
        }
    }
}

// ---------------------------------------------------------------------------
extern "C" void kernel_launch(void* const* d_in, const int* in_sizes, int n_in,
                              void* d_out, int out_size, void* d_ws, size_t ws_size,
                              hipStream_t stream) {
    const int*   nodes        = (const int*)d_in[0];
    const float* nodes_output = (const float*)d_in[1];
    const float* alpha        = (const float*)d_in[2];
    const float* item_emb     = (const float*)d_in[3];
    float* out = (float*)d_out;

    // Phase A: broadcast item_emb into all 32 graph slabs (streaming stores).
    const int per_graph4 = ITEMS * EMB / 4;  // 800000
    dim3 gridA((per_graph4 + 255) / 256, BATCH);
    broadcast_emb_kernel<<<gridA, 256, 0, stream>>>((const v4f*)item_emb, (v4f*)out);

    // Phase B: WMMA segment-mean + gated blend, overwrites touched rows.
    gated_update_kernel<<<BATCH, 512, 0, stream>>>(nodes, nodes_output, alpha,
                                                   item_emb, out);
}